// CommunicationLayer_60421599920764
// MI455X (gfx1250) — compile-verified
//
#include <hip/hip_runtime.h>

typedef __attribute__((ext_vector_type(2))) float v2f;
typedef __attribute__((ext_vector_type(4))) float v4f;
typedef __attribute__((ext_vector_type(8))) float v8f;

#define NUM_AGENTS 3
#define DDIM 128
#define BATCH 524288
#define ROWS 64                 // batch rows staged per workgroup
#define SX 132                  // padded LDS row stride in floats (bank-conflict-free frags)
#define TILE_F (ROWS * SX)      // floats per agent tile in LDS

// gfx1250 async global->LDS copy (ASYNCcnt tracked). LDS byte address is the
// low 32 bits of the generic pointer (flat->LDS aperture mapping uses addr[31:0]).
#define USE_ASYNC_STAGING 1

__global__ __launch_bounds__(256)
void CommunicationLayer_wmma_kernel(const float* __restrict__ X,
                                    const float* __restrict__ W,
                                    const float* __restrict__ bias,
                                    float* __restrict__ out)
{
    __shared__ float ldsX[NUM_AGENTS * TILE_F];   // raw X tiles, ~101 KB -> 2 WGs/WGP

    const int tid = threadIdx.x;
    const int bid = blockIdx.x;
    const size_t agentStride = (size_t)BATCH * DDIM;
    const size_t base = (size_t)bid * ROWS * DDIM;

    // ---------------- Phase 1: stage raw X tiles (3 agents) into LDS ----------------
    // 64 rows x 32 float4 per row = 2048 units, 8 per thread, async b128 DMA.
    #pragma unroll
    for (int it = 0; it < (ROWS * (DDIM / 4)) / 256; ++it) {
        const int u   = tid + it * 256;
        const int row = u >> 5;
        const int cg  = u & 31;
        const size_t g = base + (size_t)row * DDIM + (size_t)cg * 4;
        const int lo = row * SX + cg * 4;                 // floats, 16B aligned
        #pragma unroll
        for (int a = 0; a < NUM_AGENTS; ++a) {
#if USE_ASYNC_STAGING
            const unsigned lds_addr =
                (unsigned)(unsigned long long)(ldsX + a * TILE_F + lo);
            const unsigned long long gaddr =
                (unsigned long long)(X + a * agentStride + g);
            asm volatile("global_load_async_to_lds_b128 %0, %1, off"
                         :: "v"(lds_addr), "v"(gaddr) : "memory");
#else
            *(v4f*)(ldsX + a * TILE_F + lo) = *(const v4f*)(X + a * agentStride + g);
#endif
        }
    }
#if USE_ASYNC_STAGING
    asm volatile("s_wait_asynccnt 0x0" ::: "memory");   // own wave's DMAs landed in LDS
#endif
    __syncthreads();                                    // all waves' tiles visible

    // ---------------- Per-wave setup: wave owns one 16-col N-strip of W ----------------
    const int lane  = tid & 31;          // wave32
    const int wv    = tid >> 5;          // wave id 0..7 -> N-strip
    const int lhalf = lane >> 4;         // 0: lanes 0-15, 1: lanes 16-31
    const int l     = lane & 15;
    const int col   = wv * 16 + l;       // output column this lane owns (both halves)

    // B fragments for V_WMMA_F32_16X16X4_F32 (4x16 per fragment):
    //   vgpr0: lanes0-15 = W[k0+0][col], lanes16-31 = W[k0+2][col]
    //   vgpr1: lanes0-15 = W[k0+1][col], lanes16-31 = W[k0+3][col]
    // The /2 of mean-others is folded into B here (exact in fp32: exponent decrement),
    // so the inner loop A-fragment is just (xb + xc). Whole 128x16 strip stays
    // register-resident (32 x v2f = 64 VGPRs).
    const float* wp = W + (size_t)(2 * lhalf) * DDIM + col;
    v2f bfrag[32];
    #pragma unroll
    for (int ks = 0; ks < 32; ++ks) {
        v2f bf;
        bf.x = wp[(size_t)(ks * 4 + 0) * DDIM];
        bf.y = wp[(size_t)(ks * 4 + 1) * DDIM];
        bfrag[ks] = bf * 0.5f;           // one-time prescale replaces 384 inner-loop muls
    }
    const float bcol = bias[col];

    // ---------------- Phase 2: 12 tasks = 3 agents x 4 row-tiles of 16 ----------------
    // A fragment built on the fly: mean_others_a @ W = (X_b + X_c) @ (0.5*W)
    #pragma unroll 1
    for (int task = 0; task < NUM_AGENTS * (ROWS / 16); ++task) {
        const int a  = task >> 2;        // agent 0..2
        const int rt = task & 3;         // row tile 0..3
        const int ob = (a + 1) % 3;      // other agent 1
        const int oc = (a + 2) % 3;      // other agent 2

        const int fragOff = (rt * 16 + l) * SX + 2 * lhalf;
        const float* pb = ldsX + ob * TILE_F + fragOff;
        const float* pc = ldsX + oc * TILE_F + fragOff;

        v8f acc0 = {0.f, 0.f, 0.f, 0.f, 0.f, 0.f, 0.f, 0.f};
        v8f acc1 = {0.f, 0.f, 0.f, 0.f, 0.f, 0.f, 0.f, 0.f};
        #pragma unroll
        for (int ks = 0; ks < 32; ++ks) {
            const v2f xb = *(const v2f*)(pb + ks * 4);   // ds_load (2addr-merged)
            const v2f xc = *(const v2f*)(pc + ks * 4);
            const v2f af = xb + xc;                      // single v_pk_add_f32
            if (ks & 1)
                acc1 = __builtin_amdgcn_wmma_f32_16x16x4_f32(
                           false, af, false, bfrag[ks], (short)0, acc1, false, false);
            else
                acc0 = __builtin_amdgcn_wmma_f32_16x16x4_f32(
                           false, af, false, bfrag[ks], (short)0, acc0, false, false);
        }
        const v8f acc = acc0 + acc1;

        // Epilogue: out = X + bias + messages; residual X comes from LDS (conflict-free).
        const float* px = ldsX + a * TILE_F + (rt * 16) * SX;
        const size_t rowBase = (size_t)a * agentStride + base + (size_t)(rt * 16) * DDIM;
        #pragma unroll
        for (int r = 0; r < 8; ++r) {
            const int m = r + lhalf * 8;                  // C layout: vgpr r -> row r (+8)
            const float xv = px[m * SX + col];            // ds_load_b32
            out[rowBase + (size_t)m * DDIM + col] = xv + bcol + acc[r];
        }
    }
}

extern "C" void kernel_launch(void* const* d_in, const int* in_sizes, int n_in,
                              void* d_out, int out_size, void* d_ws, size_t ws_size,
                              hipStream_t stream) {
    (void)in_sizes; (void)n_in; (void)out_size; (void)d_ws; (void)ws_size;
    const float* X  = (const float*)d_in[0];   // [3, 524288, 128] fp32
    const float* W  = (const float*)d_in[1];   // [128, 128] fp32
    const float* b  = (const float*)d_in[2];   // [128] fp32
    float* out = (float*)d_out;                // [3, 524288, 128] fp32
    dim3 grid(BATCH / ROWS);                   // 8192 workgroups
    dim3 block(256);                           // 8 waves (wave32)
    CommunicationLayer_wmma_kernel<<<grid, block, 0, stream>>>(X, W, b, out);
}